// LSTM_75703093559677
// MI455X (gfx1250) — compile-verified
//
#include <hip/hip_runtime.h>

// ---------------------------------------------------------------------------
// 2-layer LSTM (B=128, T=1024, I=128, H=512) + FC on last step, MI455X/gfx1250.
// Persistent kernel, fp32 WMMA (v_wmma_f32_16x16x4_f32), grid-sync via atomics.
// Grid: 32 blocks x 256 threads (8 wave32 per block).
//   block b  -> hidden-column group nc = b   (cols [16*nc, 16*nc+16), nc in 0..31)
//   wave  w  -> batch row tile      mt = w   (rows [16*mt, 16*mt+16), mt in 0..7)
// Each wave computes the 4 gate pre-activation 16x16 tiles (i,f,g,o) for its
// (mt,nc) block in one shared-A K-loop, then does the LSTM cell update locally.
// The K-loop is explicitly software-pipelined with prefetch distance 4 chunks
// (~20 global_load_b64 in flight) so L2 latency is hidden behind the WMMAs.
// ---------------------------------------------------------------------------

#define NBLK 32
#define NTHR 256

constexpr int Bb = 128;      // batch
constexpr int Tt = 1024;     // timesteps
constexpr int Ii = 128;      // layer-0 input width
constexpr int Hh = 512;      // hidden
constexpr int BH = Bb * Hh;  // 65536 floats per state buffer (power of two)

typedef float v8f __attribute__((ext_vector_type(8)));
typedef float v2f __attribute__((ext_vector_type(2)));

__device__ __forceinline__ float fsig(float x) {
  x = fminf(15.0f, fmaxf(-15.0f, x));
  return 1.0f / (1.0f + __expf(-x));
}
__device__ __forceinline__ float ftanh(float x) {
  x = fminf(15.0f, fmaxf(-15.0f, x));
  float e = __expf(2.0f * x);
  return (e - 1.0f) / (e + 1.0f);
}

// Grid-wide barrier: monotonic arrival counter, target = barIdx * NBLK.
__device__ __forceinline__ void gridSync(unsigned* cnt, unsigned target) {
  __syncthreads();
  if (threadIdx.x == 0) {
    __threadfence();                       // release my writes (device scope)
    atomicAdd(cnt, 1u);
    volatile unsigned* vc = cnt;
    while (*vc < target) __builtin_amdgcn_s_sleep(2);
    __threadfence();                       // acquire others' writes
  }
  __syncthreads();
}

// One K=4 chunk of operands: shared A (16x4) + 4 gate B tiles (4x16).
struct Ch {
  v2f a, b0, b1, b2, b3;
};

__device__ __forceinline__ Ch ldch(const float* __restrict__ arow,
                                   const float* __restrict__ w0,
                                   const float* __restrict__ w1,
                                   const float* __restrict__ w2,
                                   const float* __restrict__ w3, int k) {
  Ch c;
  c.a  = *(const v2f*)(arow + k);
  c.b0 = *(const v2f*)(w0 + k);
  c.b1 = *(const v2f*)(w1 + k);
  c.b2 = *(const v2f*)(w2 + k);
  c.b3 = *(const v2f*)(w3 + k);
  return c;
}

__device__ __forceinline__ void domma(v8f (&acc)[4], const Ch& c) {
  acc[0] = __builtin_amdgcn_wmma_f32_16x16x4_f32(false, c.a, false, c.b0, (short)0, acc[0], false, false);
  acc[1] = __builtin_amdgcn_wmma_f32_16x16x4_f32(false, c.a, false, c.b1, (short)0, acc[1], false, false);
  acc[2] = __builtin_amdgcn_wmma_f32_16x16x4_f32(false, c.a, false, c.b2, (short)0, acc[2], false, false);
  acc[3] = __builtin_amdgcn_wmma_f32_16x16x4_f32(false, c.a, false, c.b3, (short)0, acc[3], false, false);
}

// K-segment of the 4-gate GEMM, software-pipelined with prefetch distance 4.
// All pointers are per-lane and pre-offset by 2*(lane>>4) so this lane's K pair
// sits at +k,+k+1. klen must be a multiple of 16 (chunks of K=4, 4-deep pipe).
__device__ __forceinline__ void mma4(v8f (&acc)[4], const float* __restrict__ arow,
                                     const float* __restrict__ w0,
                                     const float* __restrict__ w1,
                                     const float* __restrict__ w2,
                                     const float* __restrict__ w3, int klen) {
  Ch c0 = ldch(arow, w0, w1, w2, w3, 0);
  Ch c1 = ldch(arow, w0, w1, w2, w3, 4);
  Ch c2 = ldch(arow, w0, w1, w2, w3, 8);
  Ch c3 = ldch(arow, w0, w1, w2, w3, 12);
#pragma unroll 4
  for (int k = 0; k < klen - 16; k += 4) {
    Ch cn = ldch(arow, w0, w1, w2, w3, k + 16);
    domma(acc, c0);
    c0 = c1; c1 = c2; c2 = c3; c3 = cn;
  }
  domma(acc, c0);
  domma(acc, c1);
  domma(acc, c2);
  domma(acc, c3);
}

// Cell update for this wave's 16x16 h-block. C/D layout: element (M,N) with
// N = lane&15, M = r + 8*(lane>>4) for accumulator component r.
// State accessed as base + offset so addrspace inference keeps stores global.
__device__ __forceinline__ void cellUpdate(const v8f (&acc)[4], float* __restrict__ st,
                                           size_t cOff, size_t hOff, int mBase, int col) {
#pragma unroll
  for (int r = 0; r < 8; ++r) {
    size_t idx = (size_t)(mBase + r) * Hh + col;
    float iv = fsig(acc[0][r]);
    float fv = fsig(acc[1][r]);
    float gv = ftanh(acc[2][r]);
    float ov = fsig(acc[3][r]);
    float cn = fv * st[cOff + idx] + iv * gv;
    st[cOff + idx] = cn;
    st[hOff + idx] = ov * ftanh(cn);
  }
}

__global__ void zero_ws(float* ws, int n) {
  int i = blockIdx.x * blockDim.x + threadIdx.x;
  if (i < n) ws[i] = 0.0f;
}

__global__ __launch_bounds__(NTHR, 1) void lstm_persistent(
    const float* __restrict__ x,
    const float* __restrict__ Wih0, const float* __restrict__ Whh0,
    const float* __restrict__ bih0, const float* __restrict__ bhh0,
    const float* __restrict__ Wih1, const float* __restrict__ Whh1,
    const float* __restrict__ bih1, const float* __restrict__ bhh1,
    const float* __restrict__ fcw, const float* __restrict__ fcb,
    float* __restrict__ out, float* __restrict__ ws) {
  unsigned* syncCnt = (unsigned*)ws;  // first 64 floats reserved
  float* st = ws + 64;                // state base: h0[2] | h1[2] | c0 | c1
  const size_t H1BASE = (size_t)2 * BH;
  const size_t C0OFF  = (size_t)4 * BH;
  const size_t C1OFF  = (size_t)5 * BH;

  const int lane = threadIdx.x & 31;
  const int wv   = threadIdx.x >> 5;  // m-tile 0..7
  const int nc   = blockIdx.x;        // col group 0..31
  const int half = lane >> 4;
  const int mn   = lane & 15;
  const int mRow = wv * 16 + mn;      // A-operand row for this lane
  const int col  = nc * 16 + mn;      // B/C column for this lane
  const int koff = 2 * half;          // K-pair offset within a K=4 chunk
  const int mBase = wv * 16 + 8 * half;

  // Per-gate bias (depends on column only -> splat across accumulator rows).
  float bias0[4], bias1[4];
#pragma unroll
  for (int g = 0; g < 4; ++g) {
    bias0[g] = bih0[g * Hh + col] + bhh0[g * Hh + col];
    bias1[g] = bih1[g * Hh + col] + bhh1[g * Hh + col];
  }

  // Per-lane weight-row pointers (row = g*H + col), pre-offset by koff.
  const float *wi0[4], *wh0[4], *wi1[4], *wh1[4];
#pragma unroll
  for (int g = 0; g < 4; ++g) {
    size_t row = (size_t)(g * Hh + col);
    wi0[g] = Wih0 + row * Ii + koff;
    wh0[g] = Whh0 + row * Hh + koff;
    wi1[g] = Wih1 + row * Hh + koff;
    wh1[g] = Whh1 + row * Hh + koff;
  }

  const float* xrow = x + (size_t)mRow * Tt * Ii + koff;  // + t*Ii per step
  const size_t hRowOff = (size_t)mRow * Hh + koff;        // A rows inside h buffers
  unsigned bar = 0;

#pragma unroll 1
  for (int t = 0; t < Tt; ++t) {
    const size_t rOff = (size_t)(t & 1) * BH;  // 0 or BH
    const size_t wOff = rOff ^ (size_t)BH;     // BH or 0

    // ---------------- layer 0: [x_t | h0_prev] @ [Wih0 Whh0]^T ----------------
    v8f acc[4];
#pragma unroll
    for (int g = 0; g < 4; ++g) {
#pragma unroll
      for (int r = 0; r < 8; ++r) acc[g][r] = bias0[g];
    }
    mma4(acc, xrow + (size_t)t * Ii, wi0[0], wi0[1], wi0[2], wi0[3], Ii);
    mma4(acc, st + rOff + hRowOff,   wh0[0], wh0[1], wh0[2], wh0[3], Hh);
    cellUpdate(acc, st, C0OFF, wOff, mBase, col);
    gridSync(syncCnt, (++bar) * NBLK);

    // ---------------- layer 1: [h0_t | h1_prev] @ [Wih1 Whh1]^T ----------------
#pragma unroll
    for (int g = 0; g < 4; ++g) {
#pragma unroll
      for (int r = 0; r < 8; ++r) acc[g][r] = bias1[g];
    }
    mma4(acc, st + wOff + hRowOff,          wi1[0], wi1[1], wi1[2], wi1[3], Hh);
    mma4(acc, st + H1BASE + rOff + hRowOff, wh1[0], wh1[1], wh1[2], wh1[3], Hh);
    cellUpdate(acc, st, C1OFF, H1BASE + wOff, mBase, col);
    gridSync(syncCnt, (++bar) * NBLK);
  }

  // Final FC on the last h1. T even -> last write parity wOff = 0 -> h1[0].
  if (nc == 0 && threadIdx.x < Bb) {
    const float* hrow = st + H1BASE + ((Tt & 1) ? (size_t)BH : 0) +
                        (size_t)threadIdx.x * Hh;
    float s = fcb[0];
#pragma unroll 4
    for (int n = 0; n < Hh; ++n) s += hrow[n] * fcw[n];
    out[threadIdx.x] = s;
  }
}

extern "C" void kernel_launch(void* const* d_in, const int* in_sizes, int n_in,
                              void* d_out, int out_size, void* d_ws, size_t ws_size,
                              hipStream_t stream) {
  (void)in_sizes; (void)n_in; (void)out_size; (void)ws_size;
  const float* x    = (const float*)d_in[0];
  const float* Wih0 = (const float*)d_in[1];
  const float* Whh0 = (const float*)d_in[2];
  const float* bih0 = (const float*)d_in[3];
  const float* bhh0 = (const float*)d_in[4];
  const float* Wih1 = (const float*)d_in[5];
  const float* Whh1 = (const float*)d_in[6];
  const float* bih1 = (const float*)d_in[7];
  const float* bhh1 = (const float*)d_in[8];
  const float* fcw  = (const float*)d_in[9];
  const float* fcb  = (const float*)d_in[10];
  float* ws = (float*)d_ws;

  const int nzero = 64 + 6 * BH;  // sync counter region + h0[2], h1[2], c0, c1
  zero_ws<<<(nzero + 255) / 256, 256, 0, stream>>>(ws, nzero);
  lstm_persistent<<<NBLK, NTHR, 0, stream>>>(x, Wih0, Whh0, bih0, bhh0,
                                             Wih1, Whh1, bih1, bhh1,
                                             fcw, fcb, (float*)d_out, ws);
}